// CausalConv1d_20976620274202
// MI455X (gfx1250) — compile-verified
//
#include <hip/hip_runtime.h>
#include <stdint.h>

// Causal depthwise conv1d (K=4) + SiLU for MI455X (gfx1250, wave32).
//
// Roofline: 64M outputs, ~0.9 GFLOP vs 512 MB min traffic -> ~1.7 FLOP/B,
// pure HBM-bound; floor = 512MB / 23.3 TB/s ~ 22us. WMMA inapplicable
// (channel index appears in both factors -> not a matmul contraction), so
// the MI455X-optimal kernel is a streaming kernel on the CDNA5 async path:
// GLOBAL_LOAD_ASYNC_TO_LDS_B128 double-buffered a stage ahead (ASYNCcnt),
// register sliding window so each input byte is read exactly once, and
// non-temporal b128 stores so the 256MB streams don't thrash the 192MB L2.

typedef float v4f __attribute__((ext_vector_type(4)));
typedef int   v4i __attribute__((vector_size(16)));   // matches builtin proto

typedef __attribute__((address_space(1))) v4i as1_v4i;
typedef __attribute__((address_space(3))) v4i as3_v4i;

__device__ __forceinline__ float silu_f(float v) {
    // v * sigmoid(v) with native v_exp_f32 + v_rcp_f32
    float e = __expf(-v);
    return v * __builtin_amdgcn_rcpf(1.0f + e);
}

__device__ __forceinline__ void wait_async_le8() {
#if defined(__has_builtin)
#  if __has_builtin(__builtin_amdgcn_s_wait_asynccnt)
    __builtin_amdgcn_s_wait_asynccnt(8);
#  else
    asm volatile("s_wait_asynccnt 0x8" ::: "memory");
#  endif
#else
    asm volatile("s_wait_asynccnt 0x8" ::: "memory");
#endif
}

__device__ __forceinline__ void wait_async_le0() {
#if defined(__has_builtin)
#  if __has_builtin(__builtin_amdgcn_s_wait_asynccnt)
    __builtin_amdgcn_s_wait_asynccnt(0);
#  else
    asm volatile("s_wait_asynccnt 0x0" ::: "memory");
#  endif
#else
    asm volatile("s_wait_asynccnt 0x0" ::: "memory");
#endif
}

__global__ __launch_bounds__(256) void causal_dwconv_silu_kernel(
    const float* __restrict__ xg,   // (B, S, D)
    const float* __restrict__ wg,   // (D, 1, K) depthwise taps
    float* __restrict__ yg)         // (B, S, D)
{
    constexpr int S      = 8192;
    constexpr int D      = 2048;
    constexpr int K      = 4;
    constexpr int S_TILE = 64;

#if defined(__AMDGCN__) && defined(__has_builtin) && \
    __has_builtin(__builtin_amdgcn_global_load_async_to_lds_b128)
    // ---------------- CDNA5 async global->LDS pipeline ----------------
    constexpr int CHUNK  = 8;                // rows per stage
    constexpr int NSTAGE = S_TILE / CHUNK;   // 8 stages
    constexpr int NST    = S / S_TILE;       // 128 s-tiles

    __shared__ v4f lds_buf[2][CHUNK][256];   // 64 KB, double buffered

    const int tid   = threadIdx.x;
    const int bid   = blockIdx.x;
    const int dhalf = bid & 1;               // which 1024-channel half
    const int rest  = bid >> 1;
    const int stile = rest & (NST - 1);
    const int b     = rest >> 7;             // rest / NST
    const int d     = dhalf * 1024 + tid * 4;
    const int s0    = stile * S_TILE;

    // weights: w[(d+c)*K + k], 16 contiguous floats per thread
    const v4f* wv = reinterpret_cast<const v4f*>(wg + (size_t)d * K);
    const v4f wc0 = wv[0], wc1 = wv[1], wc2 = wv[2], wc3 = wv[3];
    const v4f t0 = {wc0.x, wc1.x, wc2.x, wc3.x};  // tap k=0 -> x[s-3]
    const v4f t1 = {wc0.y, wc1.y, wc2.y, wc3.y};  // tap k=1 -> x[s-2]
    const v4f t2 = {wc0.z, wc1.z, wc2.z, wc3.z};  // tap k=2 -> x[s-1]
    const v4f t3 = {wc0.w, wc1.w, wc2.w, wc3.w};  // tap k=3 -> x[s]

    const size_t base = ((size_t)b * S + s0) * D + d;
    const float* px = xg + base;
    float*       py = yg + base;

    // prime the K-1 sliding window (causal zero pad only for first tile)
    v4f xm3 = {0.f, 0.f, 0.f, 0.f};
    v4f xm2 = {0.f, 0.f, 0.f, 0.f};
    v4f xm1 = {0.f, 0.f, 0.f, 0.f};
    if (s0 > 0) {
        xm3 = *reinterpret_cast<const v4f*>(px - 3 * D);
        xm2 = *reinterpret_cast<const v4f*>(px - 2 * D);
        xm1 = *reinterpret_cast<const v4f*>(px - 1 * D);
    }

    // issue one stage of async copies: 8 rows x 16B/lane (b128), in-order
    auto issue_stage = [&](int stage) {
        const int buf = stage & 1;
        const float* src = px + (size_t)stage * CHUNK * D;
        #pragma unroll
        for (int k = 0; k < CHUNK; ++k) {
            __builtin_amdgcn_global_load_async_to_lds_b128(
                (as1_v4i*)(uintptr_t)(src + (size_t)k * D),
                (as3_v4i*)(uint32_t)(uintptr_t)(&lds_buf[buf][k][tid]),
                /*offset=*/0, /*cpol=*/0);
        }
    };

    issue_stage(0);
    for (int stage = 0; stage < NSTAGE; ++stage) {
        if (stage + 1 < NSTAGE) {
            issue_stage(stage + 1);   // keep next stage in flight
            wait_async_le8();         // oldest 8 (this stage) are in LDS
        } else {
            wait_async_le0();         // drain the final stage
        }
        const int buf = stage & 1;
        float* pyo = py + (size_t)stage * CHUNK * D;
        #pragma unroll
        for (int k = 0; k < CHUNK; ++k) {
            v4f x0 = lds_buf[buf][k][tid];            // ds_load_b128
            v4f acc = t0 * xm3 + t1 * xm2 + t2 * xm1 + t3 * x0;
            v4f r;
            r.x = silu_f(acc.x);
            r.y = silu_f(acc.y);
            r.z = silu_f(acc.z);
            r.w = silu_f(acc.w);
            __builtin_nontemporal_store(r,
                reinterpret_cast<v4f*>(pyo + (size_t)k * D));
            xm3 = xm2; xm2 = xm1; xm1 = x0;
        }
    }
#else
    // ---------------- fallback: register streaming pipeline ----------------
    constexpr int NDC = D / 4;         // 512 float4 channel-chunks
    constexpr int NST = S / S_TILE;    // 128 s-tiles

    const int u      = blockIdx.x * 256 + threadIdx.x;
    const int dchunk = u % NDC;
    const int t      = u / NDC;
    const int stile  = t % NST;
    const int b      = t / NST;
    const int d      = dchunk * 4;
    const int s0     = stile * S_TILE;

    const v4f* wv = reinterpret_cast<const v4f*>(wg + (size_t)d * K);
    const v4f wc0 = wv[0], wc1 = wv[1], wc2 = wv[2], wc3 = wv[3];
    const v4f t0 = {wc0.x, wc1.x, wc2.x, wc3.x};
    const v4f t1 = {wc0.y, wc1.y, wc2.y, wc3.y};
    const v4f t2 = {wc0.z, wc1.z, wc2.z, wc3.z};
    const v4f t3 = {wc0.w, wc1.w, wc2.w, wc3.w};

    const size_t base = ((size_t)b * S + s0) * D + d;
    const float* px = xg + base;
    float*       py = yg + base;

    v4f xm3 = {0.f, 0.f, 0.f, 0.f};
    v4f xm2 = {0.f, 0.f, 0.f, 0.f};
    v4f xm1 = {0.f, 0.f, 0.f, 0.f};
    if (s0 > 0) {
        xm3 = *reinterpret_cast<const v4f*>(px - 3 * D);
        xm2 = *reinterpret_cast<const v4f*>(px - 2 * D);
        xm1 = *reinterpret_cast<const v4f*>(px - 1 * D);
    }

    #pragma unroll 8
    for (int j = 0; j < S_TILE; ++j) {
        if (s0 + j + 8 < S)
            __builtin_prefetch(px + (size_t)(j + 8) * D, 0, 3);
        v4f x0 = __builtin_nontemporal_load(
                     reinterpret_cast<const v4f*>(px + (size_t)j * D));
        v4f acc = t0 * xm3 + t1 * xm2 + t2 * xm1 + t3 * x0;
        v4f r;
        r.x = silu_f(acc.x);
        r.y = silu_f(acc.y);
        r.z = silu_f(acc.z);
        r.w = silu_f(acc.w);
        __builtin_nontemporal_store(r,
            reinterpret_cast<v4f*>(py + (size_t)j * D));
        xm3 = xm2; xm2 = xm1; xm1 = x0;
    }
#endif
}

extern "C" void kernel_launch(void* const* d_in, const int* in_sizes, int n_in,
                              void* d_out, int out_size, void* d_ws, size_t ws_size,
                              hipStream_t stream) {
    (void)in_sizes; (void)n_in; (void)out_size; (void)d_ws; (void)ws_size;

    const float* x = (const float*)d_in[0];   // (4, 8192, 2048) f32
    const float* w = (const float*)d_in[1];   // (2048, 1, 4)    f32
    float*       y = (float*)d_out;           // (4, 8192, 2048) f32

    // Both bodies consume 1024 blocks x 256 threads covering B*S*D/4 v4 outputs.
    causal_dwconv_silu_kernel<<<1024, 256, 0, stream>>>(x, w, y);
}